// RNN_63720134803699
// MI455X (gfx1250) — compile-verified
//
#include <hip/hip_runtime.h>
#include <cmath>

// Elman RNN on MI455X (gfx1250), fp32 WMMA path (V_WMMA_F32_16X16X4_F32)
// + Tensor Data Mover (tensor_load_to_lds) for LDS staging.
// L=1024, B=128, D=256, H=256.

#define L_SEQ 1024
#define B_DIM 128
#define D_DIM 256
#define H_DIM 256
#define NBLK_SCAN 8

typedef __attribute__((ext_vector_type(2))) float        v2f;
typedef __attribute__((ext_vector_type(8))) float        v8f;
typedef __attribute__((ext_vector_type(4))) unsigned int v4u;
typedef __attribute__((ext_vector_type(4))) int          v4i;
typedef __attribute__((ext_vector_type(8))) int          v8i;

#define HAVE_TDM __has_builtin(__builtin_amdgcn_tensor_load_to_lds)

__device__ __forceinline__ void wait_tensorcnt0() {
#if __has_builtin(__builtin_amdgcn_s_wait_tensorcnt)
    __builtin_amdgcn_s_wait_tensorcnt(0);
#else
    asm volatile("s_wait_tensorcnt 0x0" ::: "memory");
#endif
}

#if HAVE_TDM
// 2-D f32 tile load, global -> LDS, via TDM (ISA ch.8 D# layout).
//   rows x row_elems f32 elements, row stride in elements, dense into LDS.
__device__ __forceinline__ void tdm_load_2d_f32(
    unsigned lds_byte_addr, const void* gptr,
    unsigned rows, unsigned row_elems, unsigned row_stride_elems)
{
    const unsigned long long ga = (unsigned long long)gptr;

    v4u g0;
    g0[0] = 1u;                                   // count=1, user descriptor
    g0[1] = lds_byte_addr;                        // D#.lds_addr (bytes)
    g0[2] = (unsigned)(ga & 0xFFFFFFFFu);         // global_addr[31:0]
    g0[3] = (unsigned)((ga >> 32) & 0x01FFFFFFu)  // global_addr[56:32]
          | (2u << 30);                           // type = 2 ("image")

    v8i g1;
    g1[0] = (int)(2u << 16);                      // mask=0, data_size=2 (4B)
    g1[1] = (int)((row_elems & 0xFFFFu) << 16);   // tensor_dim0[15:0]
    g1[2] = (int)(((row_elems >> 16) & 0xFFFFu)   // tensor_dim0[31:16]
          |       ((rows & 0xFFFFu) << 16));      // tensor_dim1[15:0]
    g1[3] = (int)(((rows >> 16) & 0xFFFFu)        // tensor_dim1[31:16]
          |       ((row_elems & 0xFFFFu) << 16)); // tile_dim0
    g1[4] = (int)(rows & 0xFFFFu);                // tile_dim1 (tile_dim2 = 0)
    g1[5] = (int)row_stride_elems;                // tensor_dim0_stride[31:0]
    g1[6] = 0;                                    // stride hi / dim1_stride lo
    g1[7] = 0;

    const v4i z4 = {0, 0, 0, 0};                  // 2-D: groups 2/3 unused
#if defined(__clang_major__) && (__clang_major__ >= 23)
    const v8i z8 = {0, 0, 0, 0, 0, 0, 0, 0};
    __builtin_amdgcn_tensor_load_to_lds(g0, g1, z4, z4, z8, 0);
#else
    __builtin_amdgcn_tensor_load_to_lds(g0, g1, z4, z4, 0);
#endif
}
#endif // HAVE_TDM

// Stage `rows x row_elems` f32 tile into LDS. Wave 0 drives the TDM; it waits
// on TENSORcnt before arriving at the barrier, so the barrier publishes the
// data to all waves.
template <int BLOCK>
__device__ __forceinline__ void stage_tile(
    float* lds, const float* g, unsigned rows, unsigned row_elems,
    unsigned row_stride_elems, int tid)
{
#if HAVE_TDM
    if (tid < 32) {
        tdm_load_2d_f32((unsigned)(unsigned long long)(void*)lds, (const void*)g,
                        rows, row_elems, row_stride_elems);
        wait_tensorcnt0();
    }
#else
    const float4* src = (const float4*)g;
    float4*       dst = (float4*)lds;
    for (unsigned i = tid; i < rows * row_elems / 4; i += BLOCK) dst[i] = src[i];
#endif
    __syncthreads();
}

// ---------------------------------------------------------------------------
// Kernel 1: xproj[l,b,:] = x[l,b,:] @ Wx_w^T + Wx_b  -> hAll region of d_out.
// Grid: (L*B)/16 blocks x 256 threads (8 waves). Block stages one contiguous
// 16x256 A-tile of x in LDS via TDM; each wave makes two 16x16 tiles with
// 64 fp32 WMMAs each (A from LDS, B rows of Wx_w from L2).
// ---------------------------------------------------------------------------
__global__ __launch_bounds__(256) void xproj_wmma_kernel(
    const float* __restrict__ x,      // [L*B, D]
    const float* __restrict__ Wx_w,   // [H, D]
    const float* __restrict__ Wx_b,   // [H]
    float* __restrict__ out)          // [L*B, H] (hAll region)
{
    __shared__ float atile[16 * D_DIM];          // 16 KB

    const int m0  = blockIdx.x * 16;
    const int tid = threadIdx.x;

    stage_tile<256>(atile, x + (size_t)m0 * D_DIM, 16, D_DIM, D_DIM, tid);

    const int wave = tid >> 5;
    const int lane = tid & 31;
    const int half = lane >> 4;                  // 0: lanes 0-15, 1: lanes 16-31
    const int l16  = lane & 15;
    const int kb   = half * 2;                   // K sub-offset inside 4-slice

    for (int ntile = wave; ntile < H_DIM / 16; ntile += 8) {
        const int n0 = ntile * 16;

        v8f c;
        const float bias = Wx_b[n0 + l16];
        #pragma unroll
        for (int j = 0; j < 8; ++j) c[j] = bias;

        const float* arow = atile + (size_t)l16 * D_DIM;         // LDS
        const float* brow = Wx_w + (size_t)(n0 + l16) * D_DIM;   // B[k][n]=Wx_w[n][k]

        // Pull this wave's weight rows toward L2/L0 ahead of the K-loop.
        __builtin_prefetch(brow, 0, 1);
        __builtin_prefetch(brow + D_DIM / 2, 0, 1);

        #pragma unroll 4
        for (int k0 = 0; k0 < D_DIM; k0 += 4) {
            v2f a = *(const v2f*)(arow + k0 + kb);   // (k, k+1) -> VGPR0/1
            v2f b = *(const v2f*)(brow + k0 + kb);
            c = __builtin_amdgcn_wmma_f32_16x16x4_f32(
                    false, a, false, b, (short)0, c, false, false);
        }

        // C/D layout: vgpr j -> row j + 8*half, col l16
        float* orow = out + (size_t)m0 * H_DIM + n0;
        #pragma unroll
        for (int j = 0; j < 8; ++j)
            orow[(size_t)(j + half * 8) * H_DIM + l16] = c[j];
    }
}

// ---------------------------------------------------------------------------
// Kernel 2: sequential scan  h_t = tanh(xproj_t + h_{t-1} @ Wh_w^T + Wh_b).
// 8 persistent blocks x 512 threads (16 waves). Block owns 32 columns of H;
// wave (m%8, m/8) owns one 16x16 tile of the 128x32 slice. Wh_w panel staged
// to LDS by TDM; h_{t-1} read from d_out[t-1] (L2-hot); steps separated by a
// device-scope grid barrier in d_ws.
// ---------------------------------------------------------------------------
__global__ __launch_bounds__(512) void rnn_scan_kernel(
    const float* __restrict__ h0,     // [B, H] initial hidden
    const float* __restrict__ Wh_w,   // [H, H]
    const float* __restrict__ Wh_b,   // [H]
    float* __restrict__ out,          // [L*B*H] hAll, then [B*H] h_final
    unsigned* __restrict__ bar)       // grid-barrier counter (zeroed each call)
{
    __shared__ float bpanel[32 * H_DIM];         // 32 KB: Wh_w rows n0g..n0g+31

    const int tid = threadIdx.x;
    const int n0g = blockIdx.x * 32;

    stage_tile<512>(bpanel, Wh_w + (size_t)n0g * H_DIM, 32, H_DIM, H_DIM, tid);

    const int wave = tid >> 5;
    const int lane = tid & 31;
    const int half = lane >> 4;
    const int l16  = lane & 15;
    const int kb   = half * 2;
    const int m0   = (wave & 7) * 16;            // B-row tile
    const int n0   = n0g + (wave >> 3) * 16;     // H-col tile (global)
    const int nloc = (wave >> 3) * 16 + l16;     // row inside bpanel

    const float  bias = Wh_b[n0 + l16];
    const float* bpr  = bpanel + (size_t)nloc * H_DIM;

    for (int t = 0; t < L_SEQ; ++t) {
        const float* hprev = (t == 0)
            ? h0
            : (out + (size_t)(t - 1) * B_DIM * H_DIM);
        float* obase = out + (size_t)t * B_DIM * H_DIM;

        // C init: xproj tile (already in d_out from kernel 1) + Wh_b bias.
        v8f c;
        #pragma unroll
        for (int j = 0; j < 8; ++j)
            c[j] = obase[(size_t)(m0 + j + half * 8) * H_DIM + n0 + l16] + bias;

        const float* arow = hprev + (size_t)(m0 + l16) * H_DIM;

        #pragma unroll 4
        for (int k0 = 0; k0 < H_DIM; k0 += 4) {
            v2f a = *(const v2f*)(arow + k0 + kb);   // h_{t-1}
            v2f b = *(const v2f*)(bpr  + k0 + kb);   // B[k][n] = Wh_w[n][k]
            c = __builtin_amdgcn_wmma_f32_16x16x4_f32(
                    false, a, false, b, (short)0, c, false, false);
        }

        #pragma unroll
        for (int j = 0; j < 8; ++j) {
            const float hv = tanhf(c[j]);
            const size_t idx = (size_t)(m0 + j + half * 8) * H_DIM + n0 + l16;
            obase[idx] = hv;
            if (t == L_SEQ - 1)
                out[(size_t)L_SEQ * B_DIM * H_DIM + idx] = hv;   // h_final
        }

        // -------- grid barrier: publish h_t, wait for all blocks ----------
        __syncthreads();
        if (tid == 0) {
            __threadfence();                         // release h_t writes
            atomicAdd(bar, 1u);
            const unsigned target = (unsigned)(t + 1) * NBLK_SCAN;
            while (__hip_atomic_load(bar, __ATOMIC_RELAXED,
                                     __HIP_MEMORY_SCOPE_AGENT) < target) {
                __builtin_amdgcn_s_sleep(2);
            }
        }
        __syncthreads();
        __threadfence();                             // acquire other blocks' h_t
    }
}

__global__ void init_barrier_kernel(unsigned* bar) {
    if (threadIdx.x == 0 && blockIdx.x == 0) *bar = 0u;
}

// ---------------------------------------------------------------------------
extern "C" void kernel_launch(void* const* d_in, const int* in_sizes, int n_in,
                              void* d_out, int out_size, void* d_ws, size_t ws_size,
                              hipStream_t stream) {
    const float* x    = (const float*)d_in[0];   // [L,B,D]
    const float* h0   = (const float*)d_in[1];   // [B,H]
    const float* Wx_w = (const float*)d_in[2];   // [H,D]
    const float* Wx_b = (const float*)d_in[3];   // [H]
    const float* Wh_w = (const float*)d_in[4];   // [H,H]
    const float* Wh_b = (const float*)d_in[5];   // [H]
    float* out = (float*)d_out;
    unsigned* bar = (unsigned*)d_ws;

    (void)in_sizes; (void)n_in; (void)out_size; (void)ws_size;

    // Reset grid-barrier counter (d_ws is poisoned / persists across replays).
    init_barrier_kernel<<<1, 32, 0, stream>>>(bar);

    // Phase 1: xproj for all timesteps -> hAll region of d_out.
    xproj_wmma_kernel<<<(L_SEQ * B_DIM) / 16, 256, 0, stream>>>(x, Wx_w, Wx_b, out);

    // Phase 2: sequential recurrence over L, in place on d_out.
    rnn_scan_kernel<<<NBLK_SCAN, 512, 0, stream>>>(h0, Wh_w, Wh_b, out, bar);
}